// BalancedL1Loss_4904852652721
// MI455X (gfx1250) — compile-verified
//
#include <hip/hip_runtime.h>
#include <math.h>

// ---------------------------------------------------------------------------
// BalancedL1Loss on MI455X (gfx1250), single fused streaming pass.
//
// Roofline: 134 MB read / 23.3 TB/s ~= 5.8 us floor; ~6 flops/elem => purely
// memory bound, so no WMMA (no contraction structure; building one-hot WMMA
// operands costs more than the whole arithmetic). Instead we exercise the
// CDNA5 async-tensor data path: global_load_async_to_lds_b128 (ASYNCcnt,
// s_wait_asynccnt) with double buffering, which keeps a deep memory pipeline
// without holding in-flight data in VGPRs.
//
// Pass 1 (bl1_partial): grid-stride over float4s; per element compute
//   l1 = |o-t|, slot = bin(t) in [0,15] or 16 (t < bins[0]);
//   accumulate per-slot l1-sum S and count C into 4-way replicated LDS
//   histograms (ds_add_f32); write 34 floats of per-block partials to d_ws.
// Pass 2 (bl1_final): fixed-order reduction of block partials, scalar tail,
//   EMA counts update, wi = sqrt(total/new_counts), loss = num/den.
// All reductions have fixed order/grid => bitwise deterministic.
// ---------------------------------------------------------------------------

#define BLOCK 256
#define GRIDB 2048
#define NSLOT 17   // 16 bins + 1 "invalid" (t < 0.2) slot
#define NCOPY 4    // LDS histogram replication factor

// CDNA5 async global->LDS 128b load. VDST = wave-relative LDS byte address in
// a VGPR, VADDR = 64-bit global address VGPR pair, saddr = off.
// Tracked by ASYNCcnt (CDNA5 ISA).
__device__ __forceinline__ void async_b128(unsigned lds_off,
                                           unsigned long long gaddr) {
    asm volatile("global_load_async_to_lds_b128 %0, %1, off"
                 :: "v"(lds_off), "v"(gaddr) : "memory");
}

__device__ __forceinline__ int bin_slot(float t) {
    // searchsorted(BINS, t, 'right') - 1 with BINS = 0.2 + 0.05*k.
    // floor(fma(t,20,-4)): exact at edge 0; within ~1 ULP at other edges.
    float u = fmaf(t, 20.0f, -4.0f);
    int s = (int)floorf(u);
    if (s < 0)  return 16;   // invalid slot
    if (s > 15) return 15;   // open-ended last bin
    return s;
}

__global__ __launch_bounds__(BLOCK) void bl1_partial(
    const float* __restrict__ outs, const float* __restrict__ tgts,
    float* __restrict__ partial, unsigned nvec4, unsigned gsz, unsigned iters)
{
    __shared__ float4 bufT[2][BLOCK];
    __shared__ float4 bufO[2][BLOCK];
    __shared__ float  hS[NCOPY][NSLOT];
    __shared__ float  hC[NCOPY][NSLOT];

    const unsigned tid = threadIdx.x;

    // Defined initial contents for the staging buffers + zero histograms.
    float4 z4 = make_float4(0.f, 0.f, 0.f, 0.f);
    bufT[0][tid] = z4; bufT[1][tid] = z4;
    bufO[0][tid] = z4; bufO[1][tid] = z4;
    for (unsigned i = tid; i < NCOPY * NSLOT; i += BLOCK) {
        (&hS[0][0])[i] = 0.0f;
        (&hC[0][0])[i] = 0.0f;
    }
    __syncthreads();

    const float4* t4 = (const float4*)tgts;
    const float4* o4 = (const float4*)outs;

    const unsigned gid = blockIdx.x * BLOCK + tid;   // fits 32-bit by design

    float* myS = hS[tid & (NCOPY - 1)];
    float* myC = hC[tid & (NCOPY - 1)];

    // Wave-relative LDS byte addresses (generic ptr low 32 bits = LDS offset).
    unsigned offT[2] = { (unsigned)(size_t)&bufT[0][tid],
                         (unsigned)(size_t)&bufT[1][tid] };
    unsigned offO[2] = { (unsigned)(size_t)&bufO[0][tid],
                         (unsigned)(size_t)&bufO[1][tid] };

    auto issue = [&](unsigned j) {
        unsigned idx = gid + j * gsz;
        if (idx >= nvec4) idx = nvec4 - 1;   // clamped duplicate load, masked later
        unsigned b = j & 1u;
        async_b128(offT[b], (unsigned long long)(t4 + idx));
        async_b128(offO[b], (unsigned long long)(o4 + idx));
    };

    if (iters > 0) issue(0);
    for (unsigned j = 0; j < iters; ++j) {
        if (j + 1 < iters) {
            issue(j + 1);
            // 4 outstanding async loads; <=2 left means tile j has landed
            // (async loads complete in order, ISA 08 §4.1).
            asm volatile("s_wait_asynccnt 2" ::: "memory");
        } else {
            asm volatile("s_wait_asynccnt 0" ::: "memory");
        }
        unsigned b = j & 1u;
        float4 t = bufT[b][tid];
        float4 o = bufO[b][tid];

        unsigned idx = gid + j * gsz;
        if (idx < nvec4) {
            float tv[4] = { t.x, t.y, t.z, t.w };
            float ov[4] = { o.x, o.y, o.z, o.w };
#pragma unroll
            for (int c = 0; c < 4; ++c) {
                float l1 = fabsf(ov[c] - tv[c]);
                int slot = bin_slot(tv[c]);
                atomicAdd(&myS[slot], l1);    // ds_add_f32 (no return)
                atomicAdd(&myC[slot], 1.0f);  // ds_add_f32 (no return)
            }
        }
    }
    __syncthreads();

    // Fold the 4 replicas, emit per-block partials (fixed layout: 17 S, 17 C).
    for (unsigned s = tid; s < NSLOT; s += BLOCK) {
        float S = 0.f, C = 0.f;
        for (int c = 0; c < NCOPY; ++c) { S += hS[c][s]; C += hC[c][s]; }
        unsigned base = blockIdx.x * (2 * NSLOT);
        partial[base + s]         = S;
        partial[base + NSLOT + s] = C;
    }
}

__global__ __launch_bounds__(64) void bl1_final(
    const float* __restrict__ partial, int nblocks,
    const float* __restrict__ counts_in,
    const float* __restrict__ outs, const float* __restrict__ tgts,
    unsigned n, float* __restrict__ out)
{
    __shared__ float sS[NSLOT];
    __shared__ float sC[NSLOT];
    const int tid = threadIdx.x;

    if (tid < NSLOT) {
        float S = 0.f, C = 0.f;
        for (int k = 0; k < nblocks; ++k) {          // fixed order: deterministic
            S += partial[(unsigned)k * (2 * NSLOT) + tid];
            C += partial[(unsigned)k * (2 * NSLOT) + NSLOT + tid];
        }
        sS[tid] = S; sC[tid] = C;
    }
    __syncthreads();

    if (tid == 0) {
        // Scalar tail (n not a multiple of 4).
        for (unsigned i = (n >> 2) << 2; i < n; ++i) {
            float t = tgts[i], o = outs[i];
            int slot = bin_slot(t);
            sS[slot] += fabsf(o - t);
            sC[slot] += 1.0f;
        }
        // EMA update + weights: wi = (REPEAT_THR/freq)^GAMMA = sqrt(total/nc).
        float nc[16], tot = 0.f;
        for (int b = 0; b < 16; ++b) {
            nc[b] = 0.9f * counts_in[b] + 0.1f * sC[b];
            tot += nc[b];
        }
        float num = 0.f, den = 0.f;
        for (int b = 0; b < 16; ++b) {
            float wi = sqrtf(tot / nc[b]);
            num += wi * sS[b];
            den += wi * sC[b];
        }
        num += sS[16];           // invalid elements: weight 1.0
        den += sC[16];
        out[0] = (num / den) * 1.0f;   // LOSS_WEIGHT = 1.0
    }
}

extern "C" void kernel_launch(void* const* d_in, const int* in_sizes, int n_in,
                              void* d_out, int out_size, void* d_ws, size_t ws_size,
                              hipStream_t stream) {
    (void)n_in; (void)out_size;
    const float* outs   = (const float*)d_in[0];
    const float* tgts   = (const float*)d_in[1];
    const float* counts = (const float*)d_in[2];
    unsigned n     = (unsigned)in_sizes[0];
    unsigned nvec4 = n >> 2;

    int blocks = GRIDB;
    size_t per_block = (size_t)(2 * NSLOT) * sizeof(float);
    if ((size_t)blocks * per_block > ws_size) {      // ws_size fixed per run: deterministic
        blocks = (int)(ws_size / per_block);
        if (blocks < 1) blocks = 1;
    }
    // Host-side iteration count: no 64-bit division in the kernel.
    unsigned gsz   = (unsigned)blocks * BLOCK;
    unsigned iters = nvec4 ? (nvec4 + gsz - 1) / gsz : 0;

    float* partial = (float*)d_ws;
    bl1_partial<<<blocks, BLOCK, 0, stream>>>(outs, tgts, partial, nvec4, gsz, iters);
    bl1_final<<<1, 64, 0, stream>>>(partial, blocks, counts, outs, tgts, n,
                                    (float*)d_out);
}